// RelativeAttention_58213986730531
// MI455X (gfx1250) — compile-verified
//
#include <hip/hip_runtime.h>
#include <math.h>

#define Bq  4
#define Cc  512
#define Nn  2048
#define Hh  4
#define HDd 128
#define BH  (Bq * Hh)
#define EPSc 1e-5f
#define INV_SCALE 0.08838834764831845f /* 1/sqrt(128) */
#define LDW 132   /* padded LDS row stride (floats): 132 % 64 = 4 -> conflict-free */
#define PS  17    /* padded P-tile stride */

#if defined(__has_builtin)
#if __has_builtin(__builtin_amdgcn_global_load_async_to_lds_b128) && \
    __has_builtin(__builtin_amdgcn_s_wait_asynccnt)
#define USE_ASYNC 1
#endif
#endif
#ifndef USE_ASYNC
#define USE_ASYNC 0
#endif

typedef __attribute__((ext_vector_type(2))) float v2f;
typedef __attribute__((ext_vector_type(8))) float v8f;

#if USE_ASYNC
typedef int v4i_gv __attribute__((vector_size(16)));
typedef __attribute__((address_space(1))) v4i_gv* async_gp_t;  // global
typedef __attribute__((address_space(3))) v4i_gv* async_lp_t;  // LDS
#endif

static __device__ __forceinline__ v8f wmma4(v2f a, v2f b, v8f c) {
  // D = A(16x4,f32) * B(4x16,f32) + C(16x16,f32)
  return __builtin_amdgcn_wmma_f32_16x16x4_f32(false, a, false, b, (short)0, c,
                                               false, false);
}

// Stage a 16x128 fp32 tile (row-contiguous in global) into padded LDS.
// 128 threads. Async path: 4 x global_load_async_to_lds_b128 per thread.
static __device__ __forceinline__ void tile_copy_16x128(
    float* __restrict__ dstLds, const float* __restrict__ srcGlobal, int tid) {
#if USE_ASYNC
#pragma unroll
  for (int it = 0; it < 4; ++it) {
    const int idx = (it * 128 + tid) * 4;  // float index, 16B aligned
    const int r = idx >> 7, col = idx & 127;
    __builtin_amdgcn_global_load_async_to_lds_b128(
        (async_gp_t)(srcGlobal + idx), (async_lp_t)(dstLds + r * LDW + col), 0,
        0);
  }
#else
  for (int idx = tid * 4; idx < 16 * 128; idx += 512) {
    const int r = idx >> 7, col = idx & 127;
    *(float4*)&dstLds[r * LDW + col] = *(const float4*)&srcGlobal[idx];
  }
#endif
}

// Stage a 64x128 fp32 tile into padded LDS (128 threads).
static __device__ __forceinline__ void tile_copy_64x128(
    float* __restrict__ dstLds, const float* __restrict__ srcGlobal, int tid) {
#if USE_ASYNC
#pragma unroll
  for (int it = 0; it < 16; ++it) {
    const int idx = (it * 128 + tid) * 4;
    const int r = idx >> 7, col = idx & 127;
    __builtin_amdgcn_global_load_async_to_lds_b128(
        (async_gp_t)(srcGlobal + idx), (async_lp_t)(dstLds + r * LDW + col), 0,
        0);
  }
#else
  for (int idx = tid * 4; idx < 64 * 128; idx += 512) {
    const int r = idx >> 7, col = idx & 127;
    *(float4*)&dstLds[r * LDW + col] = *(const float4*)&srcGlobal[idx];
  }
#endif
}

static __device__ __forceinline__ void async_wait() {
#if USE_ASYNC
  __builtin_amdgcn_s_wait_asynccnt(0);
#endif
}

// ---------------------------------------------------------------------------
// K1: q/k/v = W @ x  per batch; scatter-store into head layout
//     Qh[((b*H+h)*N + i)*HD + d]  with  i = 4*o + (n>>9), h=(n>>7)&3, d=n&127
// one wave per 16(o) x 64(n) tile
// ---------------------------------------------------------------------------
__global__ __launch_bounds__(32) void qkv_proj(
    const float* __restrict__ Wq, const float* __restrict__ Wk,
    const float* __restrict__ Wv, const float* __restrict__ X,
    float* __restrict__ Qh, float* __restrict__ Kh, float* __restrict__ Vh) {
  const int lane = threadIdx.x;
  const int m    = lane & 15;
  const int kp   = (lane >> 4) << 1;
  const int hi8  = (lane >> 4) << 3;
  const int n0   = blockIdx.x * 64;
  const int o0   = blockIdx.y * 16;
  const int b    = blockIdx.z / 3;
  const int which = blockIdx.z % 3;
  const float* __restrict__ W   = (which == 0) ? Wq : (which == 1) ? Wk : Wv;
  float* __restrict__       Out = (which == 0) ? Qh : (which == 1) ? Kh : Vh;
  const float* __restrict__ Xb  = X + (size_t)b * Cc * Nn;

  const v8f zero = {0.f, 0.f, 0.f, 0.f, 0.f, 0.f, 0.f, 0.f};
  v8f acc[4];
#pragma unroll
  for (int t = 0; t < 4; ++t) acc[t] = zero;

  for (int c = 0; c < Cc; c += 4) {
    v2f a = *(const v2f*)(W + (size_t)(o0 + m) * Cc + c + kp);
#pragma unroll
    for (int t = 0; t < 4; ++t) {
      const int n = n0 + t * 16 + m;
      v2f bb;
      bb.x = Xb[(size_t)(c + kp) * Nn + n];
      bb.y = Xb[(size_t)(c + kp + 1) * Nn + n];
      acc[t] = wmma4(a, bb, acc[t]);
    }
  }
#pragma unroll
  for (int t = 0; t < 4; ++t) {
#pragma unroll
    for (int r = 0; r < 8; ++r) {
      const int o = o0 + r + hi8;
      const int n = n0 + t * 16 + m;
      const int i = (o << 2) | (n >> 9);
      const int h = (n >> 7) & 3;
      const int d = n & 127;
      Out[(((size_t)(b * Hh + h)) * Nn + i) * HDd + d] = acc[t][r];
    }
  }
}

// ---------------------------------------------------------------------------
// K2: per-row softmax stats over j: rowmax/rowsum of S[n,j]/scale
// block = 4 waves; wave w owns n-rows [n0+16w, +16); K tiles double-buffered
// via async global->LDS loads overlapped with the WMMA block.
// ---------------------------------------------------------------------------
__global__ __launch_bounds__(128) void row_stats(
    const float* __restrict__ Qh, const float* __restrict__ Kh,
    float* __restrict__ rowmax, float* __restrict__ rowsum) {
  __shared__ float Qs[64 * LDW];
  __shared__ float Ks[2][16 * LDW];
  const int tid  = threadIdx.x;
  const int lane = tid & 31;
  const int w    = tid >> 5;
  const int m    = lane & 15;
  const int kp   = (lane >> 4) << 1;
  const int n0   = blockIdx.x * 64;
  const int bh   = blockIdx.y;
  const float* __restrict__ Qb = Qh + ((size_t)bh * Nn + n0) * HDd;
  const float* __restrict__ Kb = Kh + (size_t)bh * Nn * HDd;

  tile_copy_64x128(Qs, Qb, tid);
  tile_copy_16x128(Ks[0], Kb, tid);
  async_wait();
  __syncthreads();

  float rm[8], rs[8];
#pragma unroll
  for (int r = 0; r < 8; ++r) { rm[r] = -INFINITY; rs[r] = 0.f; }

  const float* qrow = &Qs[(w * 16 + m) * LDW];
  const v8f zero = {0.f, 0.f, 0.f, 0.f, 0.f, 0.f, 0.f, 0.f};

  for (int jb = 0; jb < Nn / 16; ++jb) {
    const int cur = jb & 1;
    if (jb + 1 < Nn / 16)
      tile_copy_16x128(Ks[cur ^ 1], Kb + (size_t)(jb + 1) * 16 * HDd, tid);

    v8f S = zero;
    const float* krow = &Ks[cur][m * LDW];
    for (int k = 0; k < HDd; k += 4) {
      v2f a  = *(const v2f*)&qrow[k + kp];
      v2f bb = *(const v2f*)&krow[k + kp];
      S = wmma4(a, bb, S);
    }
#pragma unroll
    for (int r = 0; r < 8; ++r) {
      float s = S[r] * INV_SCALE;
      float t = s;
      t = fmaxf(t, __shfl_xor(t, 1, 16));
      t = fmaxf(t, __shfl_xor(t, 2, 16));
      t = fmaxf(t, __shfl_xor(t, 4, 16));
      t = fmaxf(t, __shfl_xor(t, 8, 16));
      const float mnew = fmaxf(rm[r], t);
      float p = __expf(s - mnew);
      p += __shfl_xor(p, 1, 16);
      p += __shfl_xor(p, 2, 16);
      p += __shfl_xor(p, 4, 16);
      p += __shfl_xor(p, 8, 16);
      rs[r] = rs[r] * __expf(rm[r] - mnew) + p;
      rm[r] = mnew;
    }
    async_wait();
    __syncthreads();
  }
  if (m == 0) {
    const int half = lane >> 4;
#pragma unroll
    for (int r = 0; r < 8; ++r) {
      const int n = n0 + w * 16 + r + (half << 3);
      rowmax[(size_t)bh * Nn + n] = rm[r];
      rowsum[(size_t)bh * Nn + n] = rs[r];
    }
  }
}

// ---------------------------------------------------------------------------
// K3: O[m,d] = sum_n P[n,m] * V[n,d]  (P = row-softmax of S/scale)
// block = 4 waves; wave w owns m-cols [m0+16w, +16); Q/V tiles double-buffered
// via async global->LDS loads. Scatter-store O into [B,C,N].
// ---------------------------------------------------------------------------
__global__ __launch_bounds__(128) void attn_out(
    const float* __restrict__ Qh, const float* __restrict__ Kh,
    const float* __restrict__ Vh, const float* __restrict__ rowmax,
    const float* __restrict__ rowsum, float* __restrict__ Xr) {
  __shared__ float Kms[64 * LDW];
  __shared__ float Qs[2][16 * LDW];
  __shared__ float Vs[2][16 * LDW];
  __shared__ float Ps[4][16 * PS];
  const int tid  = threadIdx.x;
  const int lane = tid & 31;
  const int w    = tid >> 5;
  const int m    = lane & 15;
  const int kp   = (lane >> 4) << 1;
  const int hi8  = (lane >> 4) << 3;
  const int m0   = blockIdx.x * 64;
  const int bh   = blockIdx.y;
  const int b    = bh >> 2;
  const int h    = bh & 3;
  const float* __restrict__ Qb  = Qh + (size_t)bh * Nn * HDd;
  const float* __restrict__ Kb  = Kh + ((size_t)bh * Nn + m0) * HDd;
  const float* __restrict__ Vb  = Vh + (size_t)bh * Nn * HDd;
  const float* __restrict__ rmx = rowmax + (size_t)bh * Nn;
  const float* __restrict__ rsm = rowsum + (size_t)bh * Nn;

  tile_copy_64x128(Kms, Kb, tid);
  tile_copy_16x128(Qs[0], Qb, tid);
  tile_copy_16x128(Vs[0], Vb, tid);
  async_wait();
  __syncthreads();

  const v8f zero = {0.f, 0.f, 0.f, 0.f, 0.f, 0.f, 0.f, 0.f};
  v8f O[8];
#pragma unroll
  for (int dc = 0; dc < 8; ++dc) O[dc] = zero;

  float* Pw = Ps[w];
  const float* krow = &Kms[(w * 16 + m) * LDW];

  for (int nb = 0; nb < Nn / 16; ++nb) {
    const int cur = nb & 1;
    if (nb + 1 < Nn / 16) {
      tile_copy_16x128(Qs[cur ^ 1], Qb + (size_t)(nb + 1) * 16 * HDd, tid);
      tile_copy_16x128(Vs[cur ^ 1], Vb + (size_t)(nb + 1) * 16 * HDd, tid);
    }
    // S[nrow, mcol] for this wave's 16 m-cols
    v8f S = zero;
    for (int k = 0; k < HDd; k += 4) {
      v2f a  = *(const v2f*)&Qs[cur][m * LDW + k + kp];
      v2f bb = *(const v2f*)&krow[k + kp];
      S = wmma4(a, bb, S);
    }
    // P = exp(s - rowmax)/rowsum ; stash transposed through LDS (per-wave)
#pragma unroll
    for (int r = 0; r < 8; ++r) {
      const int nloc = r + hi8;
      const int n    = nb * 16 + nloc;
      const float p  = __expf(S[r] * INV_SCALE - rmx[n]) / rsm[n];
      Pw[nloc * PS + m] = p;
    }
    // O += P^T * V (K-dim = 16 n's, 4 wmma k-steps x 8 d-chunks)
#pragma unroll
    for (int ks = 0; ks < 4; ++ks) {
      v2f a;
      a.x = Pw[(ks * 4 + kp) * PS + m];
      a.y = Pw[(ks * 4 + kp + 1) * PS + m];
#pragma unroll
      for (int dc = 0; dc < 8; ++dc) {
        v2f bb;
        bb.x = Vs[cur][(ks * 4 + kp) * LDW + dc * 16 + m];
        bb.y = Vs[cur][(ks * 4 + kp + 1) * LDW + dc * 16 + m];
        O[dc] = wmma4(a, bb, O[dc]);
      }
    }
    async_wait();
    __syncthreads();
  }
#pragma unroll
  for (int dc = 0; dc < 8; ++dc) {
#pragma unroll
    for (int r = 0; r < 8; ++r) {
      const int mm  = m0 + w * 16 + r + hi8;
      const int d   = dc * 16 + m;
      const int cch = mm >> 2;
      const int n   = ((mm & 3) << 9) + (h << 7) + d;
      Xr[((size_t)(b * Cc + cch)) * Nn + n] = O[dc][r];
    }
  }
}

// ---------------------------------------------------------------------------
// K4: Y = Wm @ Xr, plain [B,C,N] store into d_out (pre-BN values)
// ---------------------------------------------------------------------------
__global__ __launch_bounds__(32) void out_proj(const float* __restrict__ Wm,
                                               const float* __restrict__ Xr,
                                               float* __restrict__ Y) {
  const int lane = threadIdx.x;
  const int m    = lane & 15;
  const int kp   = (lane >> 4) << 1;
  const int hi8  = (lane >> 4) << 3;
  const int n0   = blockIdx.x * 64;
  const int o0   = blockIdx.y * 16;
  const int b    = blockIdx.z;
  const float* __restrict__ Xb = Xr + (size_t)b * Cc * Nn;

  const v8f zero = {0.f, 0.f, 0.f, 0.f, 0.f, 0.f, 0.f, 0.f};
  v8f acc[4];
#pragma unroll
  for (int t = 0; t < 4; ++t) acc[t] = zero;

  for (int c = 0; c < Cc; c += 4) {
    v2f a = *(const v2f*)(Wm + (size_t)(o0 + m) * Cc + c + kp);
#pragma unroll
    for (int t = 0; t < 4; ++t) {
      const int n = n0 + t * 16 + m;
      v2f bb;
      bb.x = Xb[(size_t)(c + kp) * Nn + n];
      bb.y = Xb[(size_t)(c + kp + 1) * Nn + n];
      acc[t] = wmma4(a, bb, acc[t]);
    }
  }
#pragma unroll
  for (int t = 0; t < 4; ++t) {
#pragma unroll
    for (int r = 0; r < 8; ++r) {
      const int o = o0 + r + hi8;
      const int n = n0 + t * 16 + m;
      Y[((size_t)(b * Cc + o)) * Nn + n] = acc[t][r];
    }
  }
}

// ---------------------------------------------------------------------------
// K5/K6: BatchNorm1d training-mode stats over (B,N) per channel, then
//        normalize + affine + ReLU in place on d_out
// ---------------------------------------------------------------------------
__global__ __launch_bounds__(256) void bn_stats(const float* __restrict__ Y,
                                                float* __restrict__ stats) {
  const int c = blockIdx.x;
  float s = 0.f, sq = 0.f;
  for (int t = threadIdx.x; t < Bq * Nn; t += 256) {
    const int b = t >> 11;
    const int n = t & 2047;
    const float v = Y[((size_t)(b * Cc + c)) * Nn + n];
    s += v;
    sq += v * v;
  }
  __shared__ float sh1[256], sh2[256];
  sh1[threadIdx.x] = s;
  sh2[threadIdx.x] = sq;
  __syncthreads();
  for (int st = 128; st > 0; st >>= 1) {
    if (threadIdx.x < st) {
      sh1[threadIdx.x] += sh1[threadIdx.x + st];
      sh2[threadIdx.x] += sh2[threadIdx.x + st];
    }
    __syncthreads();
  }
  if (threadIdx.x == 0) {
    const float inv  = 1.f / (float)(Bq * Nn);
    const float mean = sh1[0] * inv;
    const float var  = sh2[0] * inv - mean * mean;
    stats[c]      = mean;
    stats[Cc + c] = rsqrtf(var + EPSc);
  }
}

__global__ __launch_bounds__(256) void bn_apply(float* __restrict__ Y,
                                                const float* __restrict__ stats,
                                                const float* __restrict__ gamma,
                                                const float* __restrict__ beta) {
  const size_t idx = (size_t)blockIdx.x * 256 + threadIdx.x;
  const int c = (int)((idx >> 11) & (Cc - 1));
  const float v =
      (Y[idx] - stats[c]) * stats[Cc + c] * gamma[c] + beta[c];
  Y[idx] = fmaxf(v, 0.f);
}

// ---------------------------------------------------------------------------
extern "C" void kernel_launch(void* const* d_in, const int* in_sizes, int n_in,
                              void* d_out, int out_size, void* d_ws,
                              size_t ws_size, hipStream_t stream) {
  (void)in_sizes; (void)n_in; (void)out_size; (void)ws_size;
  const float* x     = (const float*)d_in[0];
  /* d_in[1] = rel_pos: built but unused by the reference */
  const float* Wq    = (const float*)d_in[2];
  const float* Wk    = (const float*)d_in[3];
  const float* Wv    = (const float*)d_in[4];
  const float* Wm    = (const float*)d_in[5];
  const float* gamma = (const float*)d_in[6];
  const float* beta  = (const float*)d_in[7];
  float* Y = (float*)d_out;

  float* ws = (float*)d_ws;
  const size_t T = (size_t)Bq * Cc * Nn;  // 4,194,304 floats
  float* Qh     = ws;
  float* Kh     = Qh + T;
  float* Vh     = Kh + T;
  float* Xr     = Vh + T;
  float* rowmax = Xr + T;
  float* rowsum = rowmax + (size_t)BH * Nn;
  float* stats  = rowsum + (size_t)BH * Nn;

  qkv_proj<<<dim3(Nn / 64, Cc / 16, Bq * 3), 32, 0, stream>>>(Wq, Wk, Wv, x,
                                                              Qh, Kh, Vh);
  row_stats<<<dim3(Nn / 64, BH), 128, 0, stream>>>(Qh, Kh, rowmax, rowsum);
  attn_out<<<dim3(Nn / 64, BH), 128, 0, stream>>>(Qh, Kh, Vh, rowmax, rowsum,
                                                  Xr);
  out_proj<<<dim3(Nn / 64, Cc / 16, Bq), 32, 0, stream>>>(Wm, Xr, Y);
  bn_stats<<<Cc, 256, 0, stream>>>(Y, stats);
  bn_apply<<<(unsigned)(T / 256), 256, 0, stream>>>(Y, stats, gamma, beta);
}